// GAT_v2_Model_52690658787910
// MI455X (gfx1250) — compile-verified
//
#include <hip/hip_runtime.h>
#include <hip/hip_bf16.h>

#define N_NODES 50000
#define N_EDGES 800000
#define NHEADS  8
#define K_IN    256
#define LDSK    260   // padded row stride: 260%64=4 -> 16 distinct banks for strided A reads

typedef float v2f __attribute__((ext_vector_type(2)));
typedef float v8f __attribute__((ext_vector_type(8)));
typedef unsigned int uint32;

__device__ __forceinline__ float lrelu(float x) { return x > 0.f ? x : 0.2f * x; }

// sign-split float atomic max (global_atomic_max_i32 / global_atomic_min_u32)
__device__ __forceinline__ void atomicMaxF(float* addr, float v) {
  if (v >= 0.f) atomicMax((int*)addr, __float_as_int(v));
  else          atomicMin((unsigned int*)addr, __float_as_uint(v));
}

// ---------------------------------------------------------------------------
// H[N,M] = X[N,256] @ W[256,M] + bias[M]
// Block = 16 rows of X staged in LDS. 8 waves; each wave sweeps K once while
// accumulating TWO 16x16 output tiles (dual v8f accumulators -> 2 wmma/k-step).
// All global addressing via 32-bit element offsets (saddr + voffset form).
// A (16x4 f32): lane = (k_hi?16:0)+m, vgpr j -> K = klo+j   (v2f from LDS)
// B (4x16 f32): lane = (k_hi?16:0)+n, vgpr j -> K = klo+j   (2x b32 from global)
// C/D:          vgpr r, lanes 0-15 -> M=r, lanes 16-31 -> M=8+r, N=lane&15
// ---------------------------------------------------------------------------
__global__ __launch_bounds__(256) void gemm_wmma(const float* __restrict__ X,
                                                 const float* __restrict__ W,
                                                 const float* __restrict__ bias,
                                                 float* __restrict__ H, int M) {
  __shared__ float Xs[16 * LDSK];
  const int tid = threadIdx.x;
  const uint32 m0 = blockIdx.x * 16;

  // cooperative load: 16 rows x 256 cols, fully coalesced float4
  {
    const int r  = tid >> 4;
    const int c0 = (tid & 15) * 16;
    const float4* s = (const float4*)(X + (m0 + r) * K_IN + c0);
#pragma unroll
    for (int j = 0; j < 4; ++j)
      *(float4*)(&Xs[r * LDSK + c0 + j * 4]) = s[j];
  }
  __syncthreads();

  const int wave = tid >> 5;
  const int lane = tid & 31;
  const int mrow = lane & 15;
  const int klo  = (lane >> 4) * 2;     // 0 or 2
  const uint32 uM = (uint32)M;
  const int ntiles = M >> 4;            // 16 or 32

  for (int tmb = 0; tmb < ntiles; tmb += 16) {
    const int tm0 = tmb + wave;
    const int tm1 = tm0 + 8;
    const uint32 col0 = (uint32)tm0 * 16 + mrow;   // B/N column for tile 0
    const uint32 col1 = (uint32)tm1 * 16 + mrow;   // B/N column for tile 1
    v8f c0 = {}, c1 = {};
#pragma unroll 4
    for (int k = 0; k < K_IN; k += 4) {
      v2f a = *(const v2f*)(&Xs[mrow * LDSK + k + klo]);   // ds_load_b64
      const uint32 kk = (uint32)(k + klo) * uM;
      v2f b0, b1;
      b0.x = W[kk + col0];
      b0.y = W[kk + uM + col0];
      b1.x = W[kk + col1];
      b1.y = W[kk + uM + col1];
      c0 = __builtin_amdgcn_wmma_f32_16x16x4_f32(false, a, false, b0,
                                                 (short)0, c0, false, false);
      c1 = __builtin_amdgcn_wmma_f32_16x16x4_f32(false, a, false, b1,
                                                 (short)0, c1, false, false);
    }
    const uint32 row0 = (m0 + (uint32)(lane >> 4) * 8) * uM;
    const float bv0 = bias[col0];
    const float bv1 = bias[col1];
#pragma unroll
    for (int r = 0; r < 8; ++r) {
      H[row0 + (uint32)r * uM + col0] = c0[r] + bv0;
      H[row0 + (uint32)r * uM + col1] = c1[r] + bv1;
    }
  }
}

// ---------------------------------------------------------------------------
__global__ void fill_f32(float* __restrict__ p, float v, int n) {
  int i = blockIdx.x * blockDim.x + threadIdx.x;
  if (i < n) p[i] = v;
}

// One wave per edge. HC = 8*C floats/node; lane covers C/4 consecutive channels
// (one head: head = lane>>2). float4 gathers (L2-resident), per-head dot reduced
// over 4 lanes via shfl_xor, then segment-max atomic.
__global__ __launch_bounds__(256) void edge_logits(const float* __restrict__ h,
                                                   const long long* __restrict__ src,
                                                   const long long* __restrict__ dst,
                                                   const float* __restrict__ att,
                                                   float* __restrict__ logits,
                                                   float* __restrict__ mx,
                                                   int E, int C) {
  const int e = blockIdx.x * 8 + (threadIdx.x >> 5);
  if (e >= E) return;
  const int lane  = threadIdx.x & 31;
  const int HC    = C * NHEADS;
  const int chunk = C >> 2;          // 8 (C=32) or 16 (C=64) floats per lane
  const int head  = lane >> 2;
  const uint32 s = (uint32)src[e];
  const uint32 d = (uint32)dst[e];
  const float* hs = h + s * (uint32)HC + lane * chunk;
  const float* hd = h + d * (uint32)HC + lane * chunk;
  const float* ap = att + head * C + (lane & 3) * chunk;

  float part = 0.f;
  for (int j = 0; j < chunk; j += 4) {
    float4 a4 = *(const float4*)(hs + j);
    float4 b4 = *(const float4*)(hd + j);
    part += lrelu(a4.x + b4.x) * ap[j + 0];
    part += lrelu(a4.y + b4.y) * ap[j + 1];
    part += lrelu(a4.z + b4.z) * ap[j + 2];
    part += lrelu(a4.w + b4.w) * ap[j + 3];
  }
  part += __shfl_xor(part, 1, 32);
  part += __shfl_xor(part, 2, 32);
  if ((lane & 3) == 0) {
    logits[(uint32)e * NHEADS + head] = part;
    atomicMaxF(&mx[d * NHEADS + head], part);
  }
}

// per (edge,head): ex = exp(logit - m[dst]); denom[dst] += ex  (in-place on logits buf)
__global__ void edge_exp(const long long* __restrict__ dst,
                         const float* __restrict__ mx,
                         float* __restrict__ lg,
                         float* __restrict__ dn, int EH) {
  int i = blockIdx.x * blockDim.x + threadIdx.x;
  if (i >= EH) return;
  const int e  = i >> 3;
  const int hh = i & 7;
  const uint32 d = (uint32)dst[e];
  float m = mx[d * NHEADS + hh];
  if (!(m > -1e38f && m < 1e38f)) m = 0.f;      // nodes w/o in-edges
  float ex = expf(lg[i] - m);
  lg[i] = ex;
  atomicAdd(&dn[d * NHEADS + hh], ex);
}

// out[dst] += alpha * h[src]; one wave per edge, float4 gathers + f32 atomic scatters
__global__ __launch_bounds__(256) void aggregate(const float* __restrict__ h,
                                                 const long long* __restrict__ src,
                                                 const long long* __restrict__ dst,
                                                 const float* __restrict__ ex,
                                                 const float* __restrict__ dn,
                                                 float* __restrict__ agg,
                                                 int E, int C) {
  const int e = blockIdx.x * 8 + (threadIdx.x >> 5);
  if (e >= E) return;
  const int lane  = threadIdx.x & 31;
  const int HC    = C * NHEADS;
  const int chunk = C >> 2;
  const int head  = lane >> 2;
  const uint32 s = (uint32)src[e];
  const uint32 d = (uint32)dst[e];
  const float den   = dn[d * NHEADS + head];
  const float alpha = ex[(uint32)e * NHEADS + head] / (den > 0.f ? den : 1.f);
  const float* hs = h   + s * (uint32)HC + lane * chunk;
  float*       op = agg + d * (uint32)HC + lane * chunk;
  for (int j = 0; j < chunk; j += 4) {
    float4 v = *(const float4*)(hs + j);
    atomicAdd(op + j + 0, alpha * v.x);
    atomicAdd(op + j + 1, alpha * v.y);
    atomicAdd(op + j + 2, alpha * v.z);
    atomicAdd(op + j + 3, alpha * v.w);
  }
}

// layers 0/1 epilogue: act = elu(agg + b_out), HC == 256
__global__ void elu_bias(const float* __restrict__ agg, const float* __restrict__ bo,
                         float* __restrict__ out, int n) {
  int i = blockIdx.x * blockDim.x + threadIdx.x;
  if (i >= n) return;
  float v = agg[i] + bo[i & 255];
  out[i] = v > 0.f ? v : expf(v) - 1.f;
}

// layer 2 epilogue: mean over 8 heads of agg[N,8,64] + bias
__global__ void mean_heads(const float* __restrict__ agg, const float* __restrict__ bo,
                           float* __restrict__ out, int n) {
  int i = blockIdx.x * blockDim.x + threadIdx.x;
  if (i >= n) return;
  const uint32 nd = (uint32)i >> 6;
  const int c  = i & 63;
  const float* p = agg + nd * 512 + c;
  float s = 0.f;
#pragma unroll
  for (int hh = 0; hh < 8; ++hh) s += p[hh * 64];
  out[i] = s * 0.125f + bo[c];
}

// ---------------------------------------------------------------------------
extern "C" void kernel_launch(void* const* d_in, const int* in_sizes, int n_in,
                              void* d_out, int out_size, void* d_ws, size_t ws_size,
                              hipStream_t stream) {
  const float*     x   = (const float*)d_in[0];
  const long long* ei  = (const long long*)d_in[1];
  const long long* src = ei;
  const long long* dst = ei + N_EDGES;
  const float* Wm[3]  = {(const float*)d_in[2],  (const float*)d_in[6],  (const float*)d_in[10]};
  const float* bl[3]  = {(const float*)d_in[3],  (const float*)d_in[7],  (const float*)d_in[11]};
  const float* att[3] = {(const float*)d_in[4],  (const float*)d_in[8],  (const float*)d_in[12]};
  const float* bo[3]  = {(const float*)d_in[5],  (const float*)d_in[9],  (const float*)d_in[13]};

  float* ws  = (float*)d_ws;
  float* lin = ws;                              // [N,512] linear output h
  float* agg = lin + (size_t)N_NODES * 512;     // [N,512] aggregation
  float* act = agg + (size_t)N_NODES * 512;     // [N,256] activated (next input)
  float* exb = act + (size_t)N_NODES * 256;     // [E,8]   logits -> exp
  float* mx  = exb + (size_t)N_EDGES * 8;       // [N,8]   segment max
  float* dn  = mx  + (size_t)N_NODES * 8;       // [N,8]   segment sum

  const int Ms[3] = {256, 256, 512};
  const int NH  = N_NODES * NHEADS;
  const int EH  = N_EDGES * NHEADS;
  const float ninf = -__builtin_inff();

  const float* cur = x;
  for (int L = 0; L < 3; ++L) {
    const int M = Ms[L];
    const int C = M / NHEADS;       // 32, 32, 64
    const int NM = N_NODES * M;

    gemm_wmma<<<N_NODES / 16, 256, 0, stream>>>(cur, Wm[L], bl[L], lin, M);

    fill_f32<<<(NH + 255) / 256, 256, 0, stream>>>(mx, ninf, NH);
    fill_f32<<<(NH + 255) / 256, 256, 0, stream>>>(dn, 0.f, NH);
    fill_f32<<<(NM + 255) / 256, 256, 0, stream>>>(agg, 0.f, NM);

    edge_logits<<<N_EDGES / 8, 256, 0, stream>>>(lin, src, dst, att[L], exb, mx, N_EDGES, C);
    edge_exp<<<(EH + 255) / 256, 256, 0, stream>>>(dst, mx, exb, dn, EH);
    aggregate<<<N_EDGES / 8, 256, 0, stream>>>(lin, src, dst, exb, dn, agg, N_EDGES, C);

    if (L < 2) {
      elu_bias<<<(N_NODES * 256 + 255) / 256, 256, 0, stream>>>(agg, bo[L], act, N_NODES * 256);
      cur = act;
    } else {
      mean_heads<<<(N_NODES * 64 + 255) / 256, 256, 0, stream>>>(agg, bo[L], (float*)d_out,
                                                                 N_NODES * 64);
    }
  }
}